// GNNConv_87875030876558
// MI455X (gfx1250) — compile-verified
//
#include <hip/hip_runtime.h>
#include <hip/hip_bf16.h>

typedef float v2f __attribute__((ext_vector_type(2)));
typedef float v8f __attribute__((ext_vector_type(8)));

#define D 128

__device__ __forceinline__ v8f wmma4(v2f a, v2f b, v8f c) {
  // V_WMMA_F32_16X16X4_F32: D = A(16x4) * B(4x16) + C(16x16)
  return __builtin_amdgcn_wmma_f32_16x16x4_f32(
      /*neg_a=*/false, a, /*neg_b=*/false, b,
      /*c_mod=*/(short)0, c, /*reuse_a=*/false, /*reuse_b=*/false);
}

// ---------------------------------------------------------------------------
// Kernel 1: delta = tanh(relu(x@Wh1+bh1)@Wh2+bh2)  -> [N,3]
// One wave (32 lanes) per 16-node tile.
// A layout (16x4 f32): lane = (k>>1)*16 + m ; vgpr = k&1
// C/D layout (16x16 f32): reg r, lane L -> row = r + 8*(L>>4), col = L&15
// ---------------------------------------------------------------------------
__global__ __launch_bounds__(32) void k_delta(
    const float* __restrict__ x, const float* __restrict__ Wh1,
    const float* __restrict__ bh1, const float* __restrict__ Wh2,
    const float* __restrict__ bh2, float* __restrict__ delta, int N) {
  __shared__ float hsm[16 * D];
  const int lane = threadIdx.x;
  const int m = lane & 15;      // row for A, col for B/C
  const int kh = lane >> 4;     // which K half-pair
  const int m0 = blockIdx.x * 16;
  const int rowA = min(m0 + m, N - 1);

  v8f acc[8] = {};
  for (int k0 = 0; k0 < D; k0 += 4) {
    const int ka = k0 + 2 * kh;
    v2f a;
    a.x = x[rowA * D + ka];
    a.y = x[rowA * D + ka + 1];
#pragma unroll
    for (int t = 0; t < 8; ++t) {
      const int n0 = t * 16;
      v2f b;
      b.x = Wh1[ka * D + n0 + m];
      b.y = Wh1[(ka + 1) * D + n0 + m];
      acc[t] = wmma4(a, b, acc[t]);
    }
  }
  // bias + relu -> LDS [16][128]
#pragma unroll
  for (int t = 0; t < 8; ++t) {
    const int col = t * 16 + m;
    const float bias = bh1[col];
#pragma unroll
    for (int r = 0; r < 8; ++r) {
      const float v = acc[t][r] + bias;
      hsm[(r + 8 * kh) * D + col] = v > 0.f ? v : 0.f;
    }
  }
  __syncthreads();

  // h @ Wh2 : K=128, only 3 valid output cols (pad B with zeros)
  v8f acc2 = {};
  for (int k0 = 0; k0 < D; k0 += 4) {
    const int ka = k0 + 2 * kh;
    v2f a;
    a.x = hsm[m * D + ka];
    a.y = hsm[m * D + ka + 1];
    v2f b;
    b.x = (m < 3) ? Wh2[ka * 3 + m] : 0.f;
    b.y = (m < 3) ? Wh2[(ka + 1) * 3 + m] : 0.f;
    acc2 = wmma4(a, b, acc2);
  }
  if (m < 3) {
    const float bias = bh2[m];
#pragma unroll
    for (int r = 0; r < 8; ++r) {
      const int row = m0 + r + 8 * kh;
      if (row < N) delta[row * 3 + m] = tanhf(acc2[r] + bias);
    }
  }
}

// ---------------------------------------------------------------------------
// Kernel 0: zero accumulation buffers
// ---------------------------------------------------------------------------
__global__ void k_zero(float* __restrict__ a, long long na,
                       float* __restrict__ b, long long nb) {
  const long long tid = (long long)blockIdx.x * blockDim.x + threadIdx.x;
  const long long stride = (long long)gridDim.x * blockDim.x;
  for (long long i = tid; i < na; i += stride) a[i] = 0.f;
  for (long long i = tid; i < nb; i += stride) b[i] = 0.f;
}

// ---------------------------------------------------------------------------
// Kernel 2: edge scatter. One wave per edge iteration: 32 lanes x float4
// cover the 128 features of x[src]; lanes 0-2 handle pos, lane 0 the count.
// ---------------------------------------------------------------------------
__global__ void k_edge(const int* __restrict__ ei, const float* __restrict__ x,
                       const float* __restrict__ pos, float* __restrict__ xsum,
                       float* __restrict__ possum, float* __restrict__ counts,
                       int E) {
  const int lane = threadIdx.x & 31;
  const int wave = (int)(((long long)blockIdx.x * blockDim.x + threadIdx.x) >> 5);
  const int nwaves = (int)(((long long)gridDim.x * blockDim.x) >> 5);
  for (int e = wave; e < E; e += nwaves) {
    const int src = ei[e];        // edge_index[0][e]
    const int dst = ei[E + e];    // edge_index[1][e]
    const float4 v = ((const float4*)(x + (long long)src * D))[lane];
    float* o = xsum + (long long)dst * D + lane * 4;
    atomicAdd(o + 0, v.x);
    atomicAdd(o + 1, v.y);
    atomicAdd(o + 2, v.z);
    atomicAdd(o + 3, v.w);
    if (lane < 3) atomicAdd(possum + dst * 3 + lane, pos[src * 3 + lane]);
    if (lane == 0) atomicAdd(counts + dst, 1.0f);
  }
}

// ---------------------------------------------------------------------------
// Kernel 3: divide by counts; fold delta - pos into the pos aggregate
// ---------------------------------------------------------------------------
__global__ void k_finalize(const float* __restrict__ pos,
                           const float* __restrict__ counts,
                           float* __restrict__ possum,
                           const float* __restrict__ delta,
                           float* __restrict__ xsum, int N) {
  const long long tid = (long long)blockIdx.x * blockDim.x + threadIdx.x;
  if (tid >= (long long)N * D) return;
  const int i = (int)(tid / D);
  const int f = (int)(tid % D);
  const float inv = 1.0f / fmaxf(counts[i], 1.0f);
  xsum[(long long)i * D + f] *= inv;
  if (f < 3)
    possum[i * 3 + f] = possum[i * 3 + f] * inv + delta[i * 3 + f] - pos[i * 3 + f];
}

// ---------------------------------------------------------------------------
// Kernel 4: out = x + relu(relu(agg@Wg1+bg1)@Wg2+bg2)
// agg row i = [posagg[i] (3), xsum row i (128), 0 pad] -> K padded to 132.
// xsum lives in d_out; each tile reads its 16 rows fully before overwriting.
// ---------------------------------------------------------------------------
__global__ __launch_bounds__(32) void k_mlpg(
    const float* __restrict__ x, const float* __restrict__ posagg,
    const float* __restrict__ Wg1, const float* __restrict__ bg1,
    const float* __restrict__ Wg2, const float* __restrict__ bg2,
    float* __restrict__ out, int N) {
  __shared__ float hsm[16 * D];
  const int lane = threadIdx.x;
  const int m = lane & 15;
  const int kh = lane >> 4;
  const int m0 = blockIdx.x * 16;
  const int rowA = min(m0 + m, N - 1);

  v8f acc[8] = {};
  for (int k0 = 0; k0 < 132; k0 += 4) {
    const int ka = k0 + 2 * kh;       // even
    const int kb = ka + 1;            // odd
    v2f a;
    a.x = (ka < 3) ? posagg[rowA * 3 + ka]
                   : ((ka < 131) ? out[(long long)rowA * D + (ka - 3)] : 0.f);
    a.y = (kb < 3) ? posagg[rowA * 3 + kb]
                   : ((kb < 131) ? out[(long long)rowA * D + (kb - 3)] : 0.f);
#pragma unroll
    for (int t = 0; t < 8; ++t) {
      const int n0 = t * 16;
      v2f b;
      b.x = (ka < 131) ? Wg1[ka * D + n0 + m] : 0.f;
      b.y = (kb < 131) ? Wg1[kb * D + n0 + m] : 0.f;
      acc[t] = wmma4(a, b, acc[t]);
    }
  }
#pragma unroll
  for (int t = 0; t < 8; ++t) {
    const int col = t * 16 + m;
    const float bias = bg1[col];
#pragma unroll
    for (int r = 0; r < 8; ++r) {
      const float v = acc[t][r] + bias;
      hsm[(r + 8 * kh) * D + col] = v > 0.f ? v : 0.f;
    }
  }
  __syncthreads();

#pragma unroll
  for (int t = 0; t < 8; ++t) acc[t] = (v8f){};
  for (int k0 = 0; k0 < D; k0 += 4) {
    const int ka = k0 + 2 * kh;
    v2f a;
    a.x = hsm[m * D + ka];
    a.y = hsm[m * D + ka + 1];
#pragma unroll
    for (int t = 0; t < 8; ++t) {
      const int n0 = t * 16;
      v2f b;
      b.x = Wg2[ka * D + n0 + m];
      b.y = Wg2[(ka + 1) * D + n0 + m];
      acc[t] = wmma4(a, b, acc[t]);
    }
  }
#pragma unroll
  for (int t = 0; t < 8; ++t) {
    const int col = t * 16 + m;
    const float bias = bg2[col];
#pragma unroll
    for (int r = 0; r < 8; ++r) {
      const int row = m0 + r + 8 * kh;
      if (row < N) {
        float v = acc[t][r] + bias;
        v = v > 0.f ? v : 0.f;
        out[(long long)row * D + col] = x[(long long)row * D + col] + v;
      }
    }
  }
}

extern "C" void kernel_launch(void* const* d_in, const int* in_sizes, int n_in,
                              void* d_out, int out_size, void* d_ws, size_t ws_size,
                              hipStream_t stream) {
  const float* x   = (const float*)d_in[0];
  const float* pos = (const float*)d_in[1];
  const int*   ei  = (const int*)d_in[2];
  const float* Wh1 = (const float*)d_in[3];
  const float* bh1 = (const float*)d_in[4];
  const float* Wh2 = (const float*)d_in[5];
  const float* bh2 = (const float*)d_in[6];
  const float* Wg1 = (const float*)d_in[7];
  const float* bg1 = (const float*)d_in[8];
  const float* Wg2 = (const float*)d_in[9];
  const float* bg2 = (const float*)d_in[10];

  const int N = in_sizes[0] / D;
  const int E = in_sizes[2] / 2;
  float* out = (float*)d_out;

  // workspace layout (floats): possum[3N] | counts[N] | delta[3N]
  float* possum = (float*)d_ws;
  float* counts = possum + 3LL * N;
  float* delta  = counts + N;

  k_zero<<<4096, 256, 0, stream>>>(out, (long long)N * D, possum, 4LL * N);

  const int tiles = (N + 15) / 16;
  k_delta<<<tiles, 32, 0, stream>>>(x, Wh1, bh1, Wh2, bh2, delta, N);

  k_edge<<<2048, 256, 0, stream>>>(ei, x, pos, out, possum, counts, E);

  {
    const long long total = (long long)N * D;
    const int blocks = (int)((total + 255) / 256);
    k_finalize<<<blocks, 256, 0, stream>>>(pos, counts, possum, delta, out, N);
  }

  k_mlpg<<<tiles, 32, 0, stream>>>(x, possum, Wg1, bg1, Wg2, bg2, out, N);
}